// GAT_Three_Layer_26371099197747
// MI455X (gfx1250) — compile-verified
//
#include <hip/hip_runtime.h>
#include <math.h>

#define LRELU_SLOPE 0.2f

typedef __attribute__((ext_vector_type(2))) float v2f;
typedef __attribute__((ext_vector_type(8))) float v8f;

// ---------------------------------------------------------------------------
// GEMM: C[M,NOUT] = A[M,K] @ B[K,NOUT], f32, one wave per 16x16 tile of C.
// Uses V_WMMA_F32_16X16X4_F32 (full f32 precision).
// A 16x4 layout: lanes 0-15 -> {K=k0,k0+1}, lanes 16-31 -> {K=k0+2,k0+3}, M=lane&15
// B 4x16 layout: VGPR0 rows {k0, k0+2}, VGPR1 rows {k0+1, k0+3}, N=lane&15
// C/D: VGPR r -> M = r + 8*(lane>>4), N = lane&15
// ---------------------------------------------------------------------------
__global__ __launch_bounds__(32)
void gemm_wmma_f32(const float* __restrict__ A, const float* __restrict__ B,
                   float* __restrict__ C, int M, int K, int NOUT) {
  const int m0   = blockIdx.x << 4;
  const int n0   = blockIdx.y << 4;
  const int lane = threadIdx.x & 31;
  const int half = lane >> 4;
  const int l    = lane & 15;

  int ar = m0 + l;
  if (ar >= M) ar = M - 1;                       // clamp (keeps EXEC all-1 for WMMA)
  const float* __restrict__ arow = A + (size_t)ar * K;
  const float* __restrict__ bcol = B + n0 + l;

  v8f c = {0.f, 0.f, 0.f, 0.f, 0.f, 0.f, 0.f, 0.f};
  for (int k0 = 0; k0 < K; k0 += 4) {
    const int ka = k0 + (half << 1);
    v2f a, b;
    a.x = arow[ka];
    a.y = arow[ka + 1];
    b.x = bcol[(size_t)ka * NOUT];
    b.y = bcol[(size_t)(ka + 1) * NOUT];
    c = __builtin_amdgcn_wmma_f32_16x16x4_f32(false, a, false, b, (short)0, c,
                                              false, false);
  }
#pragma unroll
  for (int r = 0; r < 8; ++r) {
    const int row = m0 + r + (half << 3);
    if (row < M) C[(size_t)row * NOUT + n0 + l] = c[r];
  }
}

// ---------------------------------------------------------------------------
// Per-node attention logits: als[n,h] = sum_f h[n,h,f]*a_src[h,f]  (same ald)
// ---------------------------------------------------------------------------
__global__ void compute_al_kernel(const float* __restrict__ hbuf,
                                  const float* __restrict__ a_src,
                                  const float* __restrict__ a_dst,
                                  float* __restrict__ als, float* __restrict__ ald,
                                  int N, int H, int F) {
  int tid = blockIdx.x * blockDim.x + threadIdx.x;
  if (tid >= N * H) return;
  const int n = tid / H, h = tid % H;
  const float* __restrict__ hp = hbuf + (size_t)n * H * F + (size_t)h * F;
  const float* __restrict__ sp = a_src + (size_t)h * F;
  const float* __restrict__ dp = a_dst + (size_t)h * F;
  float ss = 0.f, sd = 0.f;
  for (int f = 0; f < F; ++f) {
    const float v = hp[f];
    ss += v * sp[f];
    sd += v * dp[f];
  }
  als[tid] = ss;
  ald[tid] = sd;
}

// ---------------------------------------------------------------------------
// Init accumulator / max / denom
// ---------------------------------------------------------------------------
__global__ void init_kernel(float* __restrict__ acc, float* __restrict__ mmax,
                            float* __restrict__ denom, int NFT, int NH) {
  int tid = blockIdx.x * blockDim.x + threadIdx.x;
  if (tid < NFT) acc[tid] = 0.f;
  if (tid < NH) { mmax[tid] = -3.0e38f; denom[tid] = 0.f; }
}

// ---------------------------------------------------------------------------
// Edge pass 1: segment max of leaky_relu(als[src]+ald[dst]) per (dst, head)
// self-loops: edge ids >= E map to node (e - E)
// ---------------------------------------------------------------------------
__global__ void edge_max_kernel(const int* __restrict__ src, const int* __restrict__ dst,
                                const float* __restrict__ als, const float* __restrict__ ald,
                                float* __restrict__ mmax, int E, int N, int H) {
  long long tid = (long long)blockIdx.x * blockDim.x + threadIdx.x;
  const long long total = (long long)(E + N) * H;
  if (tid >= total) return;
  const int e = (int)(tid / H);
  const int h = (int)(tid % H);
  const int s = (e < E) ? src[e] : (e - E);
  const int d = (e < E) ? dst[e] : (e - E);
  float v = als[s * H + h] + ald[d * H + h];
  v = (v > 0.f) ? v : LRELU_SLOPE * v;
  atomicMax(&mmax[d * H + h], v);
}

// ---------------------------------------------------------------------------
// Edge pass 2: compute w = exp(lrelu(.) - m[dst]) ONCE per (edge, head),
// store it, and fold the softmax denominator accumulation into the same pass.
// ---------------------------------------------------------------------------
__global__ void edge_weight_kernel(const int* __restrict__ src, const int* __restrict__ dst,
                                   const float* __restrict__ als, const float* __restrict__ ald,
                                   const float* __restrict__ mmax,
                                   float* __restrict__ wbuf, float* __restrict__ denom,
                                   int E, int N, int H) {
  long long tid = (long long)blockIdx.x * blockDim.x + threadIdx.x;
  const long long total = (long long)(E + N) * H;
  if (tid >= total) return;
  const int e = (int)(tid / H);
  const int h = (int)(tid % H);
  const int s = (e < E) ? src[e] : (e - E);
  const int d = (e < E) ? dst[e] : (e - E);
  float v = als[s * H + h] + ald[d * H + h];
  v = (v > 0.f) ? v : LRELU_SLOPE * v;
  const float w = __expf(v - mmax[d * H + h]);
  wbuf[tid] = w;
  atomicAdd(&denom[d * H + h], w);
}

// ---------------------------------------------------------------------------
// Edge pass 3: acc[dst, j0..j0+3] += w * h[src, j0..j0+3].
// One thread per 4 features -> 128-bit coalesced gather of h[src] row,
// precomputed weight broadcast from wbuf. FT=128 -> one wave32 per edge.
// ---------------------------------------------------------------------------
__global__ void edge_accum_kernel(const int* __restrict__ src, const int* __restrict__ dst,
                                  const float* __restrict__ wbuf,
                                  const float* __restrict__ hbuf,
                                  float* __restrict__ acc,
                                  int E, int N, int H, int F) {
  const int FT = H * F;
  const int Q = FT >> 2;                      // 4 features per thread
  long long tid = (long long)blockIdx.x * blockDim.x + threadIdx.x;
  const long long total = (long long)(E + N) * Q;
  if (tid >= total) return;
  const int e  = (int)(tid / Q);
  const int q  = (int)(tid % Q);
  const int j0 = q << 2;
  const int h  = j0 / F;
  const int s = (e < E) ? src[e] : (e - E);
  const int d = (e < E) ? dst[e] : (e - E);
  const float w = wbuf[(size_t)e * H + h];
  const float4 hv = *(const float4*)(hbuf + (size_t)s * FT + j0);
  float* __restrict__ ap = acc + (size_t)d * FT + j0;
  atomicAdd(ap + 0, w * hv.x);
  atomicAdd(ap + 1, w * hv.y);
  atomicAdd(ap + 2, w * hv.z);
  atomicAdd(ap + 3, w * hv.w);
}

// ---------------------------------------------------------------------------
// Finalize: out = acc/denom + bias  (optional relu)
// ---------------------------------------------------------------------------
__global__ void finalize_kernel(const float* __restrict__ acc, const float* __restrict__ denom,
                                const float* __restrict__ bias, float* __restrict__ out,
                                int N, int H, int F, int do_relu) {
  const int FT = H * F;
  int tid = blockIdx.x * blockDim.x + threadIdx.x;
  if (tid >= N * FT) return;
  const int n = tid / FT;
  const int j = tid % FT;
  const int h = j / F;
  float v = acc[tid] / denom[n * H + h] + bias[j];
  if (do_relu) v = fmaxf(v, 0.f);
  out[tid] = v;
}

// ---------------------------------------------------------------------------
// log_softmax over C=64 classes, one wave32 per node, 2 values per lane
// ---------------------------------------------------------------------------
__global__ __launch_bounds__(32)
void logsoftmax_kernel(const float* __restrict__ logits, float* __restrict__ out, int N) {
  const int node = blockIdx.x;
  if (node >= N) return;
  const int lane = threadIdx.x & 31;
  const float v0 = logits[(size_t)node * 64 + lane];
  const float v1 = logits[(size_t)node * 64 + 32 + lane];
  float mx = fmaxf(v0, v1);
  for (int off = 16; off > 0; off >>= 1) mx = fmaxf(mx, __shfl_xor(mx, off, 32));
  float s = __expf(v0 - mx) + __expf(v1 - mx);
  for (int off = 16; off > 0; off >>= 1) s += __shfl_xor(s, off, 32);
  const float lse = mx + __logf(s);
  out[(size_t)node * 64 + lane]      = v0 - lse;
  out[(size_t)node * 64 + 32 + lane] = v1 - lse;
}

// ---------------------------------------------------------------------------
extern "C" void kernel_launch(void* const* d_in, const int* in_sizes, int n_in,
                              void* d_out, int out_size, void* d_ws, size_t ws_size,
                              hipStream_t stream) {
  const int F_IN = 128, HEADS = 4, HID = 32, CLS = 64;
  const int N = in_sizes[0] / F_IN;
  const int E = in_sizes[1] / 2;

  const float* x    = (const float*)d_in[0];
  const int*   ei   = (const int*)d_in[1];
  const int*   srcv = ei;
  const int*   dstv = ei + E;
  const float* w1  = (const float*)d_in[2];
  const float* as1 = (const float*)d_in[3];
  const float* ad1 = (const float*)d_in[4];
  const float* b1  = (const float*)d_in[5];
  const float* w2  = (const float*)d_in[6];
  const float* as2 = (const float*)d_in[7];
  const float* ad2 = (const float*)d_in[8];
  const float* b2  = (const float*)d_in[9];
  const float* w3  = (const float*)d_in[10];
  const float* as3 = (const float*)d_in[11];
  const float* ad3 = (const float*)d_in[12];
  const float* b3  = (const float*)d_in[13];

  // workspace carve-out (256B aligned slices)
  char* base = (char*)d_ws;
  auto carve = [&](size_t bytes) -> float* {
    char* p = base;
    base += (bytes + 255) & ~(size_t)255;
    return (float*)p;
  };
  float* bufH   = carve((size_t)N * 128 * sizeof(float));          // gemm output h
  float* bufAcc = carve((size_t)N * 128 * sizeof(float));          // softmax numerator
  float* bufIn  = carve((size_t)N * 128 * sizeof(float));          // layer input (2,3)
  float* als    = carve((size_t)N * HEADS * sizeof(float));
  float* ald    = carve((size_t)N * HEADS * sizeof(float));
  float* mmax   = carve((size_t)N * HEADS * sizeof(float));
  float* denom  = carve((size_t)N * HEADS * sizeof(float));
  float* wbuf   = carve((size_t)(E + N) * HEADS * sizeof(float));  // per-edge weights

  const int T = 256;
  auto nb = [](long long n, int t) -> unsigned { return (unsigned)((n + t - 1) / t); };

  auto run_layer = [&](const float* in, int Kdim, const float* W,
                       const float* asrc, const float* adst, const float* bias,
                       int Hh, int Ff, float* outbuf, int do_relu) {
    const int FT = Hh * Ff;
    dim3 g((unsigned)((N + 15) / 16), (unsigned)(FT / 16));
    gemm_wmma_f32<<<g, 32, 0, stream>>>(in, W, bufH, N, Kdim, FT);
    compute_al_kernel<<<nb((long long)N * Hh, T), T, 0, stream>>>(bufH, asrc, adst,
                                                                  als, ald, N, Hh, Ff);
    init_kernel<<<nb((long long)N * FT, T), T, 0, stream>>>(bufAcc, mmax, denom,
                                                            N * FT, N * Hh);
    edge_max_kernel<<<nb((long long)(E + N) * Hh, T), T, 0, stream>>>(srcv, dstv, als, ald,
                                                                      mmax, E, N, Hh);
    edge_weight_kernel<<<nb((long long)(E + N) * Hh, T), T, 0, stream>>>(
        srcv, dstv, als, ald, mmax, wbuf, denom, E, N, Hh);
    edge_accum_kernel<<<nb((long long)(E + N) * (FT >> 2), T), T, 0, stream>>>(
        srcv, dstv, wbuf, bufH, bufAcc, E, N, Hh, Ff);
    finalize_kernel<<<nb((long long)N * FT, T), T, 0, stream>>>(bufAcc, denom, bias, outbuf,
                                                                N, Hh, Ff, do_relu);
  };

  // Layer 1: F_IN(128) -> 4 heads x 32, concat, relu
  run_layer(x, F_IN, w1, as1, ad1, b1, HEADS, HID, bufIn, 1);
  // Layer 2: 128 -> 4 heads x 32, concat, relu (in-place safe: sequential stream)
  run_layer(bufIn, HEADS * HID, w2, as2, ad2, b2, HEADS, HID, bufIn, 1);
  // Layer 3: 128 -> 1 head x 64, mean over 1 head == identity, no relu
  run_layer(bufIn, HEADS * HID, w3, as3, ad3, b3, 1, CLS, bufH, 0);

  // log_softmax over 64 classes
  logsoftmax_kernel<<<(unsigned)N, 32, 0, stream>>>(bufH, (float*)d_out, N);
}